// CoPLGCF_36000415875275
// MI455X (gfx1250) — compile-verified
//
#include <hip/hip_runtime.h>
#include <hip/hip_bf16.h>

// ---------------------------------------------------------------------------
// CDNA5 (gfx1250) implementation of the 3-layer signed-graph LightGCN encoder.
// Dense core: V_WMMA_F32_16X16X4_F32 (exact fp32 matrix path, wave32), with
// weight matrices double-buffered in LDS via GLOBAL_LOAD_ASYNC_TO_LDS_B128
// (ASYNCcnt-tracked async copy) so staging overlaps the WMMA stream.
// ---------------------------------------------------------------------------

typedef __attribute__((ext_vector_type(2))) float v2f;
typedef __attribute__((ext_vector_type(8))) float v8f;

#define N_U 100000
#define N_I 50000
#define DIM 64
#define NLAYER 3
#define NNZ_POS 2000000
#define NNZ_NEG 1000000
#define BPAIRS 8192
#define LAMBDA_REG 1e-6f
#define LEAKY 0.2f

// ---------------------------------------------------------------------------
// Zero-fill scratch / accumulators
// ---------------------------------------------------------------------------
__global__ __launch_bounds__(256) void fill0_kernel(float* __restrict__ p, long long n) {
    long long i = (long long)blockIdx.x * 256 + threadIdx.x;
    if (i < n) p[i] = 0.0f;
}

// ---------------------------------------------------------------------------
// COO SpMM via fp32 global atomics: Y[rows[e]] += vals[e] * X[cols[e]]
// One thread handles one (nnz, 4-dim) chunk -> float4 gather + 4 fp32 atomics.
// Transposed products reuse this kernel with rows/cols pointers swapped.
// ---------------------------------------------------------------------------
__global__ __launch_bounds__(256) void spmm_coo_kernel(
    const int* __restrict__ rows, const int* __restrict__ cols,
    const float* __restrict__ vals, const float* __restrict__ X,
    float* __restrict__ Y, int nnz)
{
    long long gid = (long long)blockIdx.x * 256 + threadIdx.x;
    long long total = (long long)nnz * 16;          // 16 chunks of 4 dims = 64
    if (gid >= total) return;
    int nz = (int)(gid >> 4);
    int d4 = ((int)gid & 15) * 4;
    float v = vals[nz];
    long long src = (long long)cols[nz] * DIM + d4;
    long long dst = (long long)rows[nz] * DIM + d4;
    float4 x = *(const float4*)(X + src);
    unsafeAtomicAdd(Y + dst + 0, v * x.x);          // global_atomic_add_f32
    unsafeAtomicAdd(Y + dst + 1, v * x.y);
    unsafeAtomicAdd(Y + dst + 2, v * x.z);
    unsafeAtomicAdd(Y + dst + 3, v * x.w);
}

// ---------------------------------------------------------------------------
// Async global->LDS copy of one 16B chunk per lane (ASYNCcnt tracked).
// VDST = per-lane LDS byte address (flat-addr low 32 bits == LDS offset,
// per the LDS aperture rule).
// ---------------------------------------------------------------------------
__device__ __forceinline__ void async_copy_b128(const float* g, unsigned lds_addr) {
    asm volatile("global_load_async_to_lds_b128 %0, %1, off"
                 :: "v"(lds_addr), "v"(g) : "memory");
}
__device__ __forceinline__ void wait_asynccnt_le4() {
    asm volatile("s_wait_asynccnt 0x4" ::: "memory");
}
__device__ __forceinline__ void wait_asynccnt_0() {
    asm volatile("s_wait_asynccnt 0x0" ::: "memory");
}

// ---------------------------------------------------------------------------
// One fused-linear stage: acc += A_T @ W_t^T with A_T selected at compile time
// from the register-resident tiles. 64 x v_wmma_f32_16x16x4_f32 per stage.
// A-fragment layout (ISA 7.12.2, 32-bit A 16x4): row = lane&15; lane<16 holds
// K%4 in {0,1}, lane>=16 holds K%4 in {2,3}. B (4x16): col = lane&15, same K
// split. C/D: VGPR v holds M = v + 8*(lane>>4), N = lane&15.
// ---------------------------------------------------------------------------
template <int T>
__device__ __forceinline__ void stage_compute(
    const float* __restrict__ sWbuf, int mrow, int khalf,
    const v2f* __restrict__ e2, const v2f* __restrict__ zp2,
    const v2f* __restrict__ zn2, v8f acc[4])
{
#pragma unroll
    for (int kb = 0; kb < 16; ++kb) {
        v2f a;
        if constexpr      (T == 0) a = e2[kb];
        else if constexpr (T == 1) a = zp2[kb];
        else if constexpr (T == 2) a = zp2[kb] * e2[kb];
        else if constexpr (T == 3) a = zn2[kb];
        else                       a = zn2[kb] * e2[kb];

        const int krow = kb * 4 + khalf;           // W's "in" index k
#pragma unroll
        for (int nc = 0; nc < 4; ++nc) {
            const int n = nc * 16 + mrow;          // W's "out" index n
            // B[k][n] = W[t][n][k]; the two k's per lane are contiguous.
            v2f bf = *(const v2f*)(sWbuf + n * DIM + krow);
            acc[nc] = __builtin_amdgcn_wmma_f32_16x16x4_f32(
                false, a, false, bf, (short)0, acc[nc], false, false);
        }
    }
}

// ---------------------------------------------------------------------------
// Fused 5-way linear + bias + leaky-ReLU, one layer & one side:
//   out = leaky( E@W0' + Zp@W1' + (Zp*E)@W2' + Zn@W3' + (Zn*E)@W4' + sum(b) )
// One wave (32 lanes) owns a 16-row tile; 8 waves per block share the weight
// matrices, which are double-buffered in LDS and prefetched with async
// global->LDS b128 copies while the previous stage's WMMAs run.
// ---------------------------------------------------------------------------
__global__ __launch_bounds__(256) void fused5lin_kernel(
    const float* __restrict__ E,   // [nrows,64] layer input
    const float* __restrict__ Zp,  // [nrows,64] positive aggregation
    const float* __restrict__ Zn,  // [nrows,64] negative aggregation
    const float* __restrict__ W,   // [5][64][64] row-major W[t][n][k]
    const float* __restrict__ b,   // [5][64]
    float* __restrict__ Eout,      // [nrows,64]
    int nrows)
{
    __shared__ float sW[2][DIM * DIM];   // 2 x 16 KB double buffer
    __shared__ float sB[DIM];            // summed bias

    const int lane = threadIdx.x & 31;
    const int ntiles = nrows >> 4;                 // nrows is a multiple of 16
    const int tile = blockIdx.x * 8 + (threadIdx.x >> 5);
    const bool active = tile < ntiles;
    const int tclamp = active ? tile : (ntiles - 1);

    const int mrow  = lane & 15;                   // A row / D column index
    const int khalf = (lane >> 4) * 2;             // K sub-offset per lane half
    const long long rowbase = ((long long)tclamp * 16 + mrow) * DIM;

    // Preload the three 16x64 input tiles into registers (A-fragment order).
    v2f e2[16], zp2[16], zn2[16];
#pragma unroll
    for (int kb = 0; kb < 16; ++kb) {
        long long off = rowbase + kb * 4 + khalf;
        e2[kb]  = *(const v2f*)(E  + off);
        zp2[kb] = *(const v2f*)(Zp + off);
        zn2[kb] = *(const v2f*)(Zn + off);
    }

    // Summed bias across the 5 linears (each output col gets sum_t b[t][n]).
    if (threadIdx.x < DIM) {
        float s = 0.0f;
        for (int t = 0; t < 5; ++t) s += b[t * DIM + threadIdx.x];
        sB[threadIdx.x] = s;
    }

    // Stage issuer: 256 threads x 4 chunks x 16 B = 16 KB per weight matrix;
    // 4 async b128 instructions per wave per stage (512 B/instr/wave).
    auto issue_stage = [&](int t, int buf) {
        const float* gsrc = W + t * DIM * DIM;
#pragma unroll
        for (int k = 0; k < 4; ++k) {
            int i = (int)threadIdx.x + k * 256;    // float4 chunk index 0..1023
            unsigned lds_addr = (unsigned)(unsigned long long)
                                (const void*)&sW[buf][i * 4];
            async_copy_b128(gsrc + i * 4, lds_addr);
        }
    };

    v8f acc[4] = {v8f{}, v8f{}, v8f{}, v8f{}};     // 16x64 fp32 accumulators

    // Software-pipelined stages: prefetch t+1 while computing t.
    issue_stage(0, 0);

    __syncthreads();                 // (pairs every stage: free buffer, then
    issue_stage(1, 1);               //  publish the freshly landed buffer)
    wait_asynccnt_le4();             // stage 0 resident (stage 1 in flight)
    __syncthreads();
    stage_compute<0>(sW[0], mrow, khalf, e2, zp2, zn2, acc);

    __syncthreads();
    issue_stage(2, 0);
    wait_asynccnt_le4();
    __syncthreads();
    stage_compute<1>(sW[1], mrow, khalf, e2, zp2, zn2, acc);

    __syncthreads();
    issue_stage(3, 1);
    wait_asynccnt_le4();
    __syncthreads();
    stage_compute<2>(sW[0], mrow, khalf, e2, zp2, zn2, acc);

    __syncthreads();
    issue_stage(4, 0);
    wait_asynccnt_le4();
    __syncthreads();
    stage_compute<3>(sW[1], mrow, khalf, e2, zp2, zn2, acc);

    __syncthreads();
    wait_asynccnt_0();
    __syncthreads();
    stage_compute<4>(sW[0], mrow, khalf, e2, zp2, zn2, acc);

    // Epilogue: bias + leaky ReLU + store (inactive tail waves skip stores).
    if (active) {
        const int mhalf = (lane >> 4) * 8;
#pragma unroll
        for (int nc = 0; nc < 4; ++nc) {
            const int n = nc * 16 + mrow;
            const float bias = sB[n];
#pragma unroll
            for (int v = 0; v < 8; ++v) {
                const int m = mhalf + v;
                float x = acc[nc][v] + bias;
                x = x > 0.0f ? x : LEAKY * x;
                Eout[((long long)tile * 16 + m) * DIM + n] = x;
            }
        }
    }
}

// ---------------------------------------------------------------------------
// Row-wise L2 normalize (eps 1e-12), one wave32 per 64-dim row.
// ---------------------------------------------------------------------------
__global__ __launch_bounds__(256) void l2norm_kernel(float* __restrict__ E, int nrows) {
    int row  = (int)((blockIdx.x * 256LL + threadIdx.x) >> 5);
    int lane = threadIdx.x & 31;
    if (row >= nrows) return;
    long long base = (long long)row * DIM + lane * 2;
    float2 x = *(const float2*)(E + base);
    float s = x.x * x.x + x.y * x.y;
#pragma unroll
    for (int m = 16; m > 0; m >>= 1) s += __shfl_xor(s, m, 32);
    float inv = 1.0f / fmaxf(sqrtf(s), 1e-12f);
    float2 y = {x.x * inv, x.y * inv};
    *(float2*)(E + base) = y;
}

// ---------------------------------------------------------------------------
// Gather pairs, dot-product logits, BCE-with-logits + L2 reg, atomic reduce.
// One wave32 per pair. d_out[0] = loss (pre-zeroed), d_out[1..B] = logits.
// ---------------------------------------------------------------------------
__global__ __launch_bounds__(256) void pair_loss_kernel(
    const float* __restrict__ Eu, const float* __restrict__ Ei,
    const int* __restrict__ uids, const int* __restrict__ iids,
    const float* __restrict__ labels, float* __restrict__ out)
{
    int pair = (int)((blockIdx.x * 256LL + threadIdx.x) >> 5);
    int lane = threadIdx.x & 31;
    if (pair >= BPAIRS) return;
    const float* u = Eu + (long long)uids[pair] * DIM;
    const float* w = Ei + (long long)iids[pair] * DIM;
    float2 a = *(const float2*)(u + lane * 2);
    float2 c = *(const float2*)(w + lane * 2);
    float dot = a.x * c.x + a.y * c.y;
    float uu  = a.x * a.x + a.y * a.y;
    float ii  = c.x * c.x + c.y * c.y;
#pragma unroll
    for (int m = 16; m > 0; m >>= 1) {
        dot += __shfl_xor(dot, m, 32);
        uu  += __shfl_xor(uu,  m, 32);
        ii  += __shfl_xor(ii,  m, 32);
    }
    if (lane == 0) {
        out[1 + pair] = dot;
        float y  = labels[pair];
        float sp = fmaxf(dot, 0.0f) + log1pf(expf(-fabsf(dot)));   // softplus
        float contrib = (sp - dot * y) * (1.0f / (float)BPAIRS)
                      + LAMBDA_REG * (uu + ii);
        atomicAdd(out, contrib);
    }
}

// ---------------------------------------------------------------------------
// Host-side orchestration (graph-capture safe: launches only).
// ---------------------------------------------------------------------------
extern "C" void kernel_launch(void* const* d_in, const int* in_sizes, int n_in,
                              void* d_out, int out_size, void* d_ws, size_t ws_size,
                              hipStream_t stream) {
    const int*   pos_rows = (const int*)  d_in[0];
    const int*   pos_cols = (const int*)  d_in[1];
    const float* pos_vals = (const float*)d_in[2];
    const int*   neg_rows = (const int*)  d_in[3];
    const int*   neg_cols = (const int*)  d_in[4];
    const float* neg_vals = (const float*)d_in[5];
    const int*   uids     = (const int*)  d_in[6];
    const int*   iids     = (const int*)  d_in[7];
    const float* labels   = (const float*)d_in[8];
    const float* E_u_0    = (const float*)d_in[9];
    const float* E_i_0    = (const float*)d_in[10];
    const float* W_u      = (const float*)d_in[11];  // [3][5][64][64]
    const float* b_u      = (const float*)d_in[12];  // [3][5][64]
    const float* W_i      = (const float*)d_in[13];
    const float* b_i      = (const float*)d_in[14];

    const long long NU64 = (long long)N_U * DIM;   // 6.4M floats
    const long long NI64 = (long long)N_I * DIM;   // 3.2M floats

    float* ws     = (float*)d_ws;
    float* Zu_pos = ws;                  // keep the four Z buffers contiguous
    float* Zu_neg = Zu_pos + NU64;       // so one fill0 clears them all
    float* Zi_pos = Zu_neg + NU64;
    float* Zi_neg = Zi_pos + NI64;
    float* Eu_a   = Zi_neg + NI64;
    float* Eu_b   = Eu_a + NU64;
    float* Ei_a   = Eu_b + NU64;
    float* Ei_b   = Ei_a + NI64;

    const float* src_u = E_u_0;
    const float* src_i = E_i_0;
    float* dst_u = Eu_a;
    float* dst_i = Ei_a;

    const long long zcount = 2 * NU64 + 2 * NI64;
    const int zblocks      = (int)((zcount + 255) / 256);
    const int spmm_pos_blk = (int)(((long long)NNZ_POS * 16 + 255) / 256);
    const int spmm_neg_blk = (int)(((long long)NNZ_NEG * 16 + 255) / 256);
    const int utile_blk    = (N_U / 16 + 7) / 8;   // 6250 tiles -> 782 blocks
    const int itile_blk    = (N_I / 16 + 7) / 8;   // 3125 tiles -> 391 blocks

    for (int layer = 0; layer < NLAYER; ++layer) {
        fill0_kernel<<<zblocks, 256, 0, stream>>>(Zu_pos, zcount);

        spmm_coo_kernel<<<spmm_pos_blk, 256, 0, stream>>>(
            pos_rows, pos_cols, pos_vals, src_i, Zu_pos, NNZ_POS);
        spmm_coo_kernel<<<spmm_neg_blk, 256, 0, stream>>>(
            neg_rows, neg_cols, neg_vals, src_i, Zu_neg, NNZ_NEG);
        spmm_coo_kernel<<<spmm_pos_blk, 256, 0, stream>>>(   // A^T: swap r/c
            pos_cols, pos_rows, pos_vals, src_u, Zi_pos, NNZ_POS);
        spmm_coo_kernel<<<spmm_neg_blk, 256, 0, stream>>>(
            neg_cols, neg_rows, neg_vals, src_u, Zi_neg, NNZ_NEG);

        fused5lin_kernel<<<utile_blk, 256, 0, stream>>>(
            src_u, Zu_pos, Zu_neg,
            W_u + (long long)layer * 5 * DIM * DIM,
            b_u + (long long)layer * 5 * DIM,
            dst_u, N_U);
        fused5lin_kernel<<<itile_blk, 256, 0, stream>>>(
            src_i, Zi_pos, Zi_neg,
            W_i + (long long)layer * 5 * DIM * DIM,
            b_i + (long long)layer * 5 * DIM,
            dst_i, N_I);

        src_u = dst_u;  dst_u = (dst_u == Eu_a) ? Eu_b : Eu_a;
        src_i = dst_i;  dst_i = (dst_i == Ei_a) ? Ei_b : Ei_a;
    }

    // src_u/src_i now point into workspace (layer >= 1), safe to mutate.
    l2norm_kernel<<<(int)(((long long)N_U * 32 + 255) / 256), 256, 0, stream>>>(
        (float*)src_u, N_U);

    fill0_kernel<<<1, 256, 0, stream>>>((float*)d_out, 1);   // zero loss cell
    pair_loss_kernel<<<(int)(((long long)BPAIRS * 32 + 255) / 256), 256, 0, stream>>>(
        src_u, src_i, uids, iids, labels, (float*)d_out);
}